// SinkhornKMsimple_1425929142656
// MI455X (gfx1250) — compile-verified
//
#include <hip/hip_runtime.h>

typedef float v2f __attribute__((ext_vector_type(2)));
typedef float v8f __attribute__((ext_vector_type(8)));

#define NITERS 100
#define SEPS 1e-12f

__device__ __forceinline__ float rcp_fast(float x) { return __builtin_amdgcn_rcpf(x); }
__device__ __forceinline__ float exp_fast(float x) { return __builtin_amdgcn_exp2f(x * 1.442695040888963f); }
__device__ __forceinline__ float squash(float x) {
  // 0.02 + 0.96 * sigmoid(x)
  float s = rcp_fast(1.0f + exp_fast(-x));
  return 0.02f + 0.96f * s;
}

__device__ __forceinline__ v8f wmma_k4(v2f a, v2f b, v8f c) {
  // D = A(16x4,f32) x B(4x16,f32) + C(16x16,f32)
  return __builtin_amdgcn_wmma_f32_16x16x4_f32(false, a, false, b, (short)0, c, false, false);
}

__global__ __launch_bounds__(256) void sinkhorn_km_kernel(
    const float* __restrict__ margins,
    const float* __restrict__ W1, const float* __restrict__ b1,
    const float* __restrict__ W2, const float* __restrict__ b2,
    const float* __restrict__ W3, const float* __restrict__ b3,
    const float* __restrict__ Wtau,
    float* __restrict__ out_mus, float* __restrict__ out_V,
    int nrows)
{
  __shared__ float s_marg[256][12];   // per-row margins
  __shared__ float s_h1[8][16][32];   // per-wave layer-1 activations (one 16-row tile)
  __shared__ float s_h2[8][16][16];   // per-wave layer-2 activations
  __shared__ float s_pars[256][18];   // per-row MLP outputs

  const int tid  = threadIdx.x;
  const int w    = tid >> 5;          // wave id in block (8 waves)
  const int lane = tid & 31;
  const int half = lane >> 4;         // 0: lanes 0-15, 1: lanes 16-31
  const int lq   = lane & 15;

  const int row  = blockIdx.x * 256 + tid;
  const int lrow = (row < nrows) ? row : (nrows - 1); // clamp for safe loads

  // ---- stage this lane's 12 margins into LDS (rows are 48B -> float4 aligned) ----
  {
    const float4* mrow = (const float4*)(margins + (size_t)lrow * 12);
    float4* sm = (float4*)&s_marg[tid][0];
    sm[0] = mrow[0];
    sm[1] = mrow[1];
    sm[2] = mrow[2];
  }
  // Same-wave DS ops are processed in order; producers == consumers per wave,
  // and per-wave LDS regions are disjoint, so no block barrier is required.

  // ================= MLP via V_WMMA_F32_16X16X4_F32 =================
  // Two 16-row tiles per wave.
  #pragma unroll
  for (int t = 0; t < 2; ++t) {
    const int rbase = w * 32 + t * 16;

    // ---- layer 1: (16x12) @ W1(12x32) + b1, ReLU ----
    #pragma unroll
    for (int nt = 0; nt < 2; ++nt) {
      v8f acc = {0.f,0.f,0.f,0.f,0.f,0.f,0.f,0.f};
      const int col = nt * 16 + lq;
      #pragma unroll
      for (int kc = 0; kc < 3; ++kc) {
        const int k0 = kc * 4 + half * 2;
        v2f a; a.x = s_marg[rbase + lq][k0];
               a.y = s_marg[rbase + lq][k0 + 1];
        v2f b; b.x = W1[(k0    ) * 32 + col];
               b.y = W1[(k0 + 1) * 32 + col];
        acc = wmma_k4(a, b, acc);
      }
      const float bias = b1[col];
      #pragma unroll
      for (int vi = 0; vi < 8; ++vi) {
        float h = acc[vi] + bias;
        s_h1[w][vi + 8 * half][col] = (h > 0.f) ? h : 0.f;
      }
    }

    // ---- layer 2: (16x32) @ W2(32x16) + b2, ReLU ----
    {
      v8f acc = {0.f,0.f,0.f,0.f,0.f,0.f,0.f,0.f};
      #pragma unroll
      for (int kc = 0; kc < 8; ++kc) {
        const int k0 = kc * 4 + half * 2;
        v2f a; a.x = s_h1[w][lq][k0];
               a.y = s_h1[w][lq][k0 + 1];
        v2f b; b.x = W2[(k0    ) * 16 + lq];
               b.y = W2[(k0 + 1) * 16 + lq];
        acc = wmma_k4(a, b, acc);
      }
      const float bias = b2[lq];
      #pragma unroll
      for (int vi = 0; vi < 8; ++vi) {
        float h = acc[vi] + bias;
        s_h2[w][vi + 8 * half][lq] = (h > 0.f) ? h : 0.f;
      }
    }

    // ---- layer 3: (16x16) @ W3(16x18) + b3 (N padded 18 -> 32) ----
    #pragma unroll
    for (int nt = 0; nt < 2; ++nt) {
      v8f acc = {0.f,0.f,0.f,0.f,0.f,0.f,0.f,0.f};
      const int col   = nt * 16 + lq;
      const bool valid = (col < 18);
      #pragma unroll
      for (int kc = 0; kc < 4; ++kc) {
        const int k0 = kc * 4 + half * 2;
        v2f a; a.x = s_h2[w][lq][k0];
               a.y = s_h2[w][lq][k0 + 1];
        v2f b; b.x = valid ? W3[(k0    ) * 18 + col] : 0.f;
               b.y = valid ? W3[(k0 + 1) * 18 + col] : 0.f;
        acc = wmma_k4(a, b, acc);
      }
      if (valid) {
        const float bias = b3[col];
        #pragma unroll
        for (int vi = 0; vi < 8; ++vi)
          s_pars[rbase + vi + 8 * half][col] = acc[vi] + bias;
      }
    }
  }

  // ================= per-row Sinkhorn (thread-per-row, registers) =================
  float p[18];
  #pragma unroll
  for (int k = 0; k < 18; ++k) p[k] = s_pars[tid][k];

  // A = exp(einsum pars[0:8] . Wtau) — Wtau addresses are wave-uniform (scalar loads)
  float Amat[6][6];
  #pragma unroll
  for (int i = 0; i < 6; ++i) {
    #pragma unroll
    for (int j = 0; j < 6; ++j) {
      float acc = 0.f;
      #pragma unroll
      for (int k = 0; k < 8; ++k) acc += p[k] * Wtau[k * 36 + i * 6 + j];
      Amat[i][j] = exp_fast(acc);
    }
  }

  float Mv[6], Fv[6];
  #pragma unroll
  for (int i = 0; i < 6; ++i) { Mv[i] = s_marg[tid][i]; Fv[i] = s_marg[tid][6 + i]; }

  float shm[6], shf[6];
  shm[0] = squash(p[9]);  shm[1] = squash(p[10]); shm[2] = 1.f;
  shm[3] = squash(p[11]); shm[4] = squash(p[12]); shm[5] = 1.f;
  shf[0] = squash(p[13]); shf[1] = squash(p[14]); shf[2] = 1.f;
  shf[3] = squash(p[15]); shf[4] = squash(p[16]); shf[5] = 1.f;
  const float Vout = exp_fast(p[17]);

  float rm[6], cm[6], mum[6], muf[6];
  #pragma unroll
  for (int i = 0; i < 6; ++i) {
    rm[i]  = Mv[i] * shm[i];
    mum[i] = Mv[i] * (1.f - shm[i]);
    cm[i]  = Fv[i] * shf[i];
    muf[i] = Fv[i] * (1.f - shf[i]);
  }

  for (int it = 0; it < NITERS; ++it) {
    // row normalization toward rm
    #pragma unroll
    for (int i = 0; i < 6; ++i) {
      float s = Amat[i][0] + Amat[i][1] + Amat[i][2] + Amat[i][3] + Amat[i][4] + Amat[i][5];
      float sc = rm[i] * rcp_fast(s + SEPS);
      #pragma unroll
      for (int j = 0; j < 6; ++j) Amat[i][j] *= sc;
    }
    // column normalization toward cm
    #pragma unroll
    for (int j = 0; j < 6; ++j) {
      float s = Amat[0][j] + Amat[1][j] + Amat[2][j] + Amat[3][j] + Amat[4][j] + Amat[5][j];
      float sc = cm[j] * rcp_fast(s + SEPS);
      #pragma unroll
      for (int i = 0; i < 6; ++i) Amat[i][j] *= sc;
    }
  }

  // ================= write outputs =================
  if (row < nrows) {
    float* o = out_mus + (size_t)row * 49;
    #pragma unroll
    for (int i = 0; i < 6; ++i) {
      #pragma unroll
      for (int j = 0; j < 6; ++j) o[i * 7 + j] = Amat[i][j];
      o[i * 7 + 6] = mum[i];
    }
    #pragma unroll
    for (int j = 0; j < 6; ++j) o[42 + j] = muf[j];
    o[48] = 0.f;
    out_V[row] = Vout;
  }
}

extern "C" void kernel_launch(void* const* d_in, const int* in_sizes, int n_in,
                              void* d_out, int out_size, void* d_ws, size_t ws_size,
                              hipStream_t stream) {
  (void)n_in; (void)out_size; (void)d_ws; (void)ws_size;
  const float* margins = (const float*)d_in[0];
  const float* W1   = (const float*)d_in[1];
  const float* b1   = (const float*)d_in[2];
  const float* W2   = (const float*)d_in[3];
  const float* b2   = (const float*)d_in[4];
  const float* W3   = (const float*)d_in[5];
  const float* b3   = (const float*)d_in[6];
  const float* Wtau = (const float*)d_in[7];

  const int nrows = in_sizes[0] / 12;
  float* out_mus = (float*)d_out;
  float* out_V   = out_mus + (size_t)nrows * 49;

  const int blocks = (nrows + 255) / 256;
  hipLaunchKernelGGL(sinkhorn_km_kernel, dim3(blocks), dim3(256), 0, stream,
                     margins, W1, b1, W2, b2, W3, b3, Wtau, out_mus, out_V, nrows);
}